// MultiHeadGAT_38010460569730
// MI455X (gfx1250) — compile-verified
//
#include <hip/hip_runtime.h>
#include <hip/hip_bf16.h>
#include <math.h>

typedef __attribute__((ext_vector_type(2))) float v2f;
typedef __attribute__((ext_vector_type(8))) float v8f;

#define GAT_N   3072
#define GAT_F   512
#define GAT_H   8
#define GAT_D   64
#define NEG_INF (-9e16f)

// ---------------------------------------------------------------------------
// Kernel 1: ht[h][n][d] = sum_f h[n][f] * W[h][f][d]
// One wave per 16x16 (n,d) tile, f32 WMMA 16x16x4, K-loop over 512.
// Tiles: 8 heads * 192 n-tiles * 4 d-tiles = 6144 -> 768 blocks * 8 waves.
// ---------------------------------------------------------------------------
__global__ __launch_bounds__(256) void gat_proj_kernel(
    const float* __restrict__ h, const float* __restrict__ W,
    float* __restrict__ ht) {
  const int lane = threadIdx.x & 31;
  const int wave = threadIdx.x >> 5;
  const int tile = blockIdx.x * 8 + wave;      // 0..6143
  const int dt   = tile & 3;                   // d-tile
  const int rest = tile >> 2;
  const int nt   = rest % (GAT_N / 16);        // n-tile
  const int hh   = rest / (GAT_N / 16);        // head
  const int nbase = nt * 16;
  const int dbase = dt * 16;
  const int col  = lane & 15;                  // matrix row (A) / col (B,C)
  const int half = lane >> 4;                  // selects K-pair

  v8f c = {};
  // A: lane holds h[nbase+col][k0 + half*2 + {0,1}]  (aligned float2)
  const float* Ap = h + (size_t)(nbase + col) * GAT_F + half * 2;
  // B: lane holds W[hh][k0 + half*2 + {0,1}][dbase+col] (stride-D apart)
  const float* Bp = W + ((size_t)hh * GAT_F + half * 2) * GAT_D + dbase + col;

  for (int k0 = 0; k0 < GAT_F; k0 += 4) {
    v2f a = *(const v2f*)(Ap + k0);
    v2f b;
    b.x = Bp[(size_t)k0 * GAT_D];
    b.y = Bp[(size_t)k0 * GAT_D + GAT_D];
    c = __builtin_amdgcn_wmma_f32_16x16x4_f32(false, a, false, b,
                                              (short)0, c, false, false);
  }

  // C layout: VGPR r -> M = r + half*8, N = col
  float* Cp = ht + ((size_t)hh * GAT_N + nbase + half * 8) * GAT_D + dbase + col;
#pragma unroll
  for (int r = 0; r < 8; ++r) Cp[(size_t)r * GAT_D] = c[r];
}

// ---------------------------------------------------------------------------
// Kernel 2: src[h][n] = ht[h][n][:]·a[h][0:64],  dst[h][n] = ht[h][n][:]·a[h][64:128]
// One thread per (h, n).
// ---------------------------------------------------------------------------
__global__ __launch_bounds__(256) void gat_attvec_kernel(
    const float* __restrict__ ht, const float* __restrict__ a,
    float* __restrict__ srcb, float* __restrict__ dstb) {
  const int idx = blockIdx.x * 256 + threadIdx.x;      // 0..24575
  const int hh = idx / GAT_N;
  const float* hp = ht + (size_t)idx * GAT_D;
  const float* as = a + (size_t)hh * 2 * GAT_D;
  float s = 0.f, d = 0.f;
#pragma unroll 8
  for (int k = 0; k < GAT_D; ++k) {
    float v = hp[k];
    s += v * as[k];
    d += v * as[GAT_D + k];
  }
  srcb[idx] = s;
  dstb[idx] = d;
}

// ---------------------------------------------------------------------------
// Kernel 3: per wave = one (head, 16-row n-tile).
//  Pass 1: online softmax stats (max, sum) over m -- no stores.
//  Pass 2: recompute e, write normalized att (float2/lane, coalesced),
//          feed p directly into WMMA A-operand; accumulate hp (16x64) in
//          4 x v8f; B operands (ht rows) stream from L2.
// Tiles: 8 * 192 = 1536 -> 384 blocks * 4 waves.
// ---------------------------------------------------------------------------
__global__ __launch_bounds__(128) void gat_attn_kernel(
    const float* __restrict__ ht, const int* __restrict__ adj,
    const float* __restrict__ srcb, const float* __restrict__ dstb,
    float* __restrict__ out, float* __restrict__ att) {
  const int lane = threadIdx.x & 31;
  const int wave = threadIdx.x >> 5;
  const int tile = blockIdx.x * 4 + wave;      // 0..1535
  const int nt = tile % (GAT_N / 16);
  const int hh = tile / (GAT_N / 16);
  const int nbase = nt * 16;
  const int row  = lane & 15;                  // row of A / col of B,C
  const int half = lane >> 4;
  const int n = nbase + row;

  const float sv = srcb[hh * GAT_N + n];
  const int*   adjp = adj  + (size_t)n * GAT_N + half * 2;
  const float* dstp = dstb + (size_t)hh * GAT_N + half * 2;

  // ---- pass 1: online max / sum over this lane's half of the row ----
  float mx = -INFINITY, sum = 0.f;
  for (int m0 = 0; m0 < GAT_N; m0 += 4) {
    int2   ad = *(const int2*)(adjp + m0);
    float2 dv = *(const float2*)(dstp + m0);
    float e0 = sv + dv.x; e0 = e0 > 0.f ? e0 : 0.2f * e0; if (ad.x <= 0) e0 = NEG_INF;
    float e1 = sv + dv.y; e1 = e1 > 0.f ? e1 : 0.2f * e1; if (ad.y <= 0) e1 = NEG_INF;
    float mn = fmaxf(mx, fmaxf(e0, e1));
    sum = sum * __expf(mx - mn) + __expf(e0 - mn) + __expf(e1 - mn);
    mx = mn;
  }
  // merge the two half-lanes of each row (lane l <-> lane l^16), wave32 shuffle
  {
    float mx2 = __shfl_xor(mx, 16, 32);
    float s2  = __shfl_xor(sum, 16, 32);
    float mn = fmaxf(mx, mx2);
    sum = sum * __expf(mx - mn) + s2 * __expf(mx2 - mn);
    mx = mn;
  }
  const float inv = 1.0f / sum;

  // ---- pass 2: write att, accumulate hp via WMMA ----
  v8f c0 = {}, c1 = {}, c2 = {}, c3 = {};
  float* attp = att + ((size_t)hh * GAT_N + n) * GAT_N + half * 2;
  // B: lane holds ht[hh][m0 + half*2 + {0,1}][dt*16 + row]
  const float* Bp = ht + ((size_t)hh * GAT_N + half * 2) * GAT_D + row;

  for (int m0 = 0; m0 < GAT_N; m0 += 4) {
    int2   ad = *(const int2*)(adjp + m0);
    float2 dv = *(const float2*)(dstp + m0);
    float e0 = sv + dv.x; e0 = e0 > 0.f ? e0 : 0.2f * e0; if (ad.x <= 0) e0 = NEG_INF;
    float e1 = sv + dv.y; e1 = e1 > 0.f ? e1 : 0.2f * e1; if (ad.y <= 0) e1 = NEG_INF;
    float p0 = __expf(e0 - mx) * inv;
    float p1 = __expf(e1 - mx) * inv;
    *(float2*)(attp + m0) = make_float2(p0, p1);

    v2f a; a.x = p0; a.y = p1;
    const float* bp = Bp + (size_t)m0 * GAT_D;
    v2f b0, b1, b2, b3;
    b0.x = bp[0];  b0.y = bp[GAT_D + 0];
    b1.x = bp[16]; b1.y = bp[GAT_D + 16];
    b2.x = bp[32]; b2.y = bp[GAT_D + 32];
    b3.x = bp[48]; b3.y = bp[GAT_D + 48];
    c0 = __builtin_amdgcn_wmma_f32_16x16x4_f32(false, a, false, b0, (short)0, c0, false, false);
    c1 = __builtin_amdgcn_wmma_f32_16x16x4_f32(false, a, false, b1, (short)0, c1, false, false);
    c2 = __builtin_amdgcn_wmma_f32_16x16x4_f32(false, a, false, b2, (short)0, c2, false, false);
    c3 = __builtin_amdgcn_wmma_f32_16x16x4_f32(false, a, false, b3, (short)0, c3, false, false);
  }

  // out[n'][hh*64 + d]: C layout VGPR r -> M = r + half*8, N = row
  float* op = out + (size_t)(nbase + half * 8) * (GAT_H * GAT_D) + hh * GAT_D + row;
#pragma unroll
  for (int r = 0; r < 8; ++r) {
    float* o = op + (size_t)r * (GAT_H * GAT_D);
    o[0]  = c0[r];
    o[16] = c1[r];
    o[32] = c2[r];
    o[48] = c3[r];
  }
}

// ---------------------------------------------------------------------------
extern "C" void kernel_launch(void* const* d_in, const int* in_sizes, int n_in,
                              void* d_out, int out_size, void* d_ws, size_t ws_size,
                              hipStream_t stream) {
  const float* h   = (const float*)d_in[0];   // (3072, 512)
  const int*   adj = (const int*)d_in[1];     // (3072, 3072)
  const float* W   = (const float*)d_in[2];   // (8, 512, 64)
  const float* a   = (const float*)d_in[3];   // (8, 128, 1)

  float* out = (float*)d_out;                           // (3072, 512)
  float* att = (float*)d_out + (size_t)GAT_N * GAT_H * GAT_D;  // (8, 3072, 3072)

  float* ht   = (float*)d_ws;                           // H*N*D
  float* srcb = ht + (size_t)GAT_H * GAT_N * GAT_D;     // H*N
  float* dstb = srcb + (size_t)GAT_H * GAT_N;           // H*N

  // K1: 6144 wave-tiles / 8 waves per block
  gat_proj_kernel<<<768, 256, 0, stream>>>(h, W, ht);
  // K2: 24576 threads
  gat_attvec_kernel<<<(GAT_H * GAT_N) / 256, 256, 0, stream>>>(ht, a, srcb, dstb);
  // K3: 1536 wave-tiles / 4 waves per block
  gat_attn_kernel<<<384, 128, 0, stream>>>(ht, adj, srcb, dstb, out, att);
}